// GCNII_57148834840952
// MI455X (gfx1250) — compile-verified
//
#include <hip/hip_runtime.h>
#include <math.h>

// ---------------- problem constants (match reference) ----------------
#define NNODES   100000
#define NEDGES   1600000
#define FIN      512
#define HDIM     64
#define NCLS     40
#define NLAYERS  4
#define ALPHA    0.1f

typedef float v2f __attribute__((ext_vector_type(2)));
typedef float v8f __attribute__((ext_vector_type(8)));

// ---------------- graph normalization ----------------

// deg[i] = 1.0 (self loop weight); will later hold dinv in place
__global__ __launch_bounds__(256) void k_fill_deg(float* __restrict__ deg) {
    int i = blockIdx.x * 256 + threadIdx.x;
    if (i < NNODES) deg[i] = 1.0f;
}

__global__ __launch_bounds__(256) void k_edge_deg(const int* __restrict__ dst,
                                                  const float* __restrict__ w,
                                                  float* __restrict__ deg) {
    int e = blockIdx.x * 256 + threadIdx.x;
    if (e < NEDGES) unsafeAtomicAdd(&deg[dst[e]], w[e]);
}

__global__ __launch_bounds__(256) void k_dinv(float* __restrict__ deg) {
    int i = blockIdx.x * 256 + threadIdx.x;
    if (i < NNODES) {
        float d = deg[i];
        deg[i] = (d > 0.0f) ? rsqrtf(d) : 0.0f;
    }
}

__global__ __launch_bounds__(256) void k_norm(const int* __restrict__ src,
                                              const int* __restrict__ dst,
                                              const float* __restrict__ w,
                                              const float* __restrict__ dinv,
                                              float* __restrict__ norm) {
    int e = blockIdx.x * 256 + threadIdx.x;
    if (e < NEDGES) norm[e] = dinv[src[e]] * w[e] * dinv[dst[e]];
}

// ---------------- input projection: h = relu(x @ W_in + b), h0 = h ----------------
// One wave computes a 16x64 output strip using V_WMMA_F32_16X16X4_F32.
// A layout (16x4 f32): lane L, vgpr v -> M = L%16, K = 2*(L/16) + v
// B layout (4x16 f32): lane L, vgpr v -> N = L%16, K = 2*(L/16) + v
// C/D layout:          lane L, vgpr v -> N = L%16, M = v + 8*(L/16)
__global__ __launch_bounds__(256) void k_input_gemm(const float* __restrict__ x,
                                                    const float* __restrict__ Win,
                                                    const float* __restrict__ bin,
                                                    float* __restrict__ h,
                                                    float* __restrict__ h0) {
    const int lane = threadIdx.x & 31;
    const int wave = threadIdx.x >> 5;
    const int tile = blockIdx.x * 8 + wave;
    if (tile * 16 >= NNODES) return;
    const int m0   = tile * 16;
    const int lm   = lane & 15;
    const int half = lane >> 4;

    v8f acc0 = {}, acc1 = {}, acc2 = {}, acc3 = {};
    const float* arow = x + (size_t)(m0 + lm) * FIN + half * 2;

    for (int kb = 0; kb < FIN / 4; ++kb) {
        v2f a = *(const v2f*)(arow + kb * 4);
        const int k0 = kb * 4 + half * 2;
        const float* Bp = Win + (size_t)k0 * HDIM + lm;
        v2f b0 = { Bp[0],  Bp[HDIM + 0]  };
        v2f b1 = { Bp[16], Bp[HDIM + 16] };
        v2f b2 = { Bp[32], Bp[HDIM + 32] };
        v2f b3 = { Bp[48], Bp[HDIM + 48] };
        acc0 = __builtin_amdgcn_wmma_f32_16x16x4_f32(false, a, false, b0, (short)0, acc0, false, false);
        acc1 = __builtin_amdgcn_wmma_f32_16x16x4_f32(false, a, false, b1, (short)0, acc1, false, false);
        acc2 = __builtin_amdgcn_wmma_f32_16x16x4_f32(false, a, false, b2, (short)0, acc2, false, false);
        acc3 = __builtin_amdgcn_wmma_f32_16x16x4_f32(false, a, false, b3, (short)0, acc3, false, false);
    }

    #pragma unroll
    for (int v = 0; v < 8; ++v) {
        const int row = m0 + half * 8 + v;
        float* hp  = h  + (size_t)row * HDIM;
        float* h0p = h0 + (size_t)row * HDIM;
        float r0 = fmaxf(acc0[v] + bin[lm +  0], 0.0f);
        float r1 = fmaxf(acc1[v] + bin[lm + 16], 0.0f);
        float r2 = fmaxf(acc2[v] + bin[lm + 32], 0.0f);
        float r3 = fmaxf(acc3[v] + bin[lm + 48], 0.0f);
        hp[lm +  0] = r0; h0p[lm +  0] = r0;
        hp[lm + 16] = r1; h0p[lm + 16] = r1;
        hp[lm + 32] = r2; h0p[lm + 32] = r2;
        hp[lm + 48] = r3; h0p[lm + 48] = r3;
    }
}

// ---------------- per-layer sparse propagation ----------------

// agg[i,:] = dinv[i]^2 * h[i,:]   (self-loop contribution)
__global__ __launch_bounds__(256) void k_agg_init(const float* __restrict__ h,
                                                  const float* __restrict__ dinv,
                                                  float* __restrict__ agg) {
    int idx = blockIdx.x * 256 + threadIdx.x;
    if (idx < NNODES * HDIM) {
        float di = dinv[idx >> 6];
        agg[idx] = di * di * h[idx];
    }
}

// one wave per edge; lane l handles features 2l, 2l+1
__global__ __launch_bounds__(256) void k_spmm(const int* __restrict__ src,
                                              const int* __restrict__ dst,
                                              const float* __restrict__ norm,
                                              const float* __restrict__ h,
                                              float* __restrict__ agg) {
    int gid = blockIdx.x * 256 + threadIdx.x;
    int e = gid >> 5;
    if (e >= NEDGES) return;
    int lane = gid & 31;
    int s = src[e], d = dst[e];
    float nv = norm[e];
    v2f hv = *(const v2f*)(h + (size_t)s * HDIM + lane * 2);
    float* ap = agg + (size_t)d * HDIM + lane * 2;
    unsafeAtomicAdd(ap + 0, nv * hv.x);
    unsafeAtomicAdd(ap + 1, nv * hv.y);
}

// ---------------- per-layer dense update ----------------
// support = (1-a)*agg + a*h0 ; h = relu(beta*(support@W) + (1-beta)*support)
__global__ __launch_bounds__(256) void k_layer_gemm(const float* __restrict__ agg,
                                                    const float* __restrict__ h0,
                                                    const float* __restrict__ W,
                                                    float* __restrict__ h,
                                                    float beta) {
    __shared__ float Ws[HDIM * HDIM];   // 16 KB
    for (int i = threadIdx.x; i < HDIM * HDIM; i += 256) Ws[i] = W[i];
    __syncthreads();

    const int lane = threadIdx.x & 31;
    const int wave = threadIdx.x >> 5;
    const int tile = blockIdx.x * 8 + wave;
    if (tile * 16 >= NNODES) return;
    const int m0   = tile * 16;
    const int lm   = lane & 15;
    const int half = lane >> 4;

    v8f acc0 = {}, acc1 = {}, acc2 = {}, acc3 = {};
    const size_t rbase = (size_t)(m0 + lm) * HDIM + half * 2;

    for (int kb = 0; kb < HDIM / 4; ++kb) {
        v2f ga = *(const v2f*)(agg + rbase + kb * 4);
        v2f g0 = *(const v2f*)(h0  + rbase + kb * 4);
        v2f a  = { (1.0f - ALPHA) * ga.x + ALPHA * g0.x,
                   (1.0f - ALPHA) * ga.y + ALPHA * g0.y };
        const int k0 = kb * 4 + half * 2;
        const float* Bp = Ws + k0 * HDIM + lm;
        v2f b0 = { Bp[0],  Bp[HDIM + 0]  };
        v2f b1 = { Bp[16], Bp[HDIM + 16] };
        v2f b2 = { Bp[32], Bp[HDIM + 32] };
        v2f b3 = { Bp[48], Bp[HDIM + 48] };
        acc0 = __builtin_amdgcn_wmma_f32_16x16x4_f32(false, a, false, b0, (short)0, acc0, false, false);
        acc1 = __builtin_amdgcn_wmma_f32_16x16x4_f32(false, a, false, b1, (short)0, acc1, false, false);
        acc2 = __builtin_amdgcn_wmma_f32_16x16x4_f32(false, a, false, b2, (short)0, acc2, false, false);
        ac3:;
        acc3 = __builtin_amdgcn_wmma_f32_16x16x4_f32(false, a, false, b3, (short)0, acc3, false, false);
    }

    const float omb = 1.0f - beta;
    #pragma unroll
    for (int v = 0; v < 8; ++v) {
        const int row = m0 + half * 8 + v;
        const size_t rb = (size_t)row * HDIM;
        #pragma unroll
        for (int nt = 0; nt < 4; ++nt) {
            const int col = nt * 16 + lm;
            float sup = (1.0f - ALPHA) * agg[rb + col] + ALPHA * h0[rb + col];
            float d = (nt == 0) ? acc0[v] : (nt == 1) ? acc1[v] : (nt == 2) ? acc2[v] : acc3[v];
            h[rb + col] = fmaxf(beta * d + omb * sup, 0.0f);
        }
    }
}

// ---------------- output projection + log_softmax ----------------
__global__ __launch_bounds__(256) void k_out(const float* __restrict__ h,
                                             const float* __restrict__ Wout,
                                             const float* __restrict__ bout,
                                             float* __restrict__ out) {
    __shared__ float hs[64 * HDIM];        // 16 KB
    __shared__ float Ws[HDIM * NCLS];      // 10 KB
    __shared__ float bs[NCLS];
    __shared__ float lg[64 * NCLS];        // 10 KB

    const int r0 = blockIdx.x * 64;
    const int nrows = (NNODES - r0 < 64) ? (NNODES - r0) : 64;

    for (int i = threadIdx.x; i < HDIM * NCLS; i += 256) Ws[i] = Wout[i];
    if (threadIdx.x < NCLS) bs[threadIdx.x] = bout[threadIdx.x];
    for (int i = threadIdx.x; i < nrows * HDIM; i += 256) hs[i] = h[(size_t)r0 * HDIM + i];
    __syncthreads();

    for (int idx = threadIdx.x; idx < nrows * NCLS; idx += 256) {
        int r = idx / NCLS, c = idx % NCLS;
        float s = bs[c];
        #pragma unroll 8
        for (int k = 0; k < HDIM; ++k) s += hs[r * HDIM + k] * Ws[k * NCLS + c];
        lg[idx] = s;
    }
    __syncthreads();

    if (threadIdx.x < nrows) {
        const int r = threadIdx.x;
        float m = -1e30f;
        for (int c = 0; c < NCLS; ++c) m = fmaxf(m, lg[r * NCLS + c]);
        float sum = 0.0f;
        for (int c = 0; c < NCLS; ++c) sum += __expf(lg[r * NCLS + c] - m);
        float ls = m + __logf(sum);
        float* op = out + (size_t)(r0 + r) * NCLS;
        for (int c = 0; c < NCLS; ++c) op[c] = lg[r * NCLS + c] - ls;
    }
}

// ---------------- host orchestration ----------------
extern "C" void kernel_launch(void* const* d_in, const int* in_sizes, int n_in,
                              void* d_out, int out_size, void* d_ws, size_t ws_size,
                              hipStream_t stream) {
    const float* x    = (const float*)d_in[0];
    const int*   ei   = (const int*)  d_in[1];
    const float* ew   = (const float*)d_in[2];
    const float* Win  = (const float*)d_in[3];
    const float* bin  = (const float*)d_in[4];
    const float* Wl   = (const float*)d_in[5];
    const float* Wout = (const float*)d_in[6];
    const float* bout = (const float*)d_in[7];
    float* out = (float*)d_out;

    const int* esrc = ei;
    const int* edst = ei + NEDGES;

    float* ws   = (float*)d_ws;
    float* h    = ws;
    float* h0   = h   + (size_t)NNODES * HDIM;
    float* agg  = h0  + (size_t)NNODES * HDIM;
    float* dinv = agg + (size_t)NNODES * HDIM;
    float* nrm  = dinv + NNODES;

    const int nodeBlocks = (NNODES + 255) / 256;
    const int edgeBlocks = (NEDGES + 255) / 256;
    const int tileBlocks = (NNODES / 16 + 7) / 8;   // 8 waves (tiles) per 256-thread block

    k_fill_deg<<<nodeBlocks, 256, 0, stream>>>(dinv);
    k_edge_deg<<<edgeBlocks, 256, 0, stream>>>(edst, ew, dinv);
    k_dinv    <<<nodeBlocks, 256, 0, stream>>>(dinv);
    k_norm    <<<edgeBlocks, 256, 0, stream>>>(esrc, edst, ew, dinv, nrm);

    k_input_gemm<<<tileBlocks, 256, 0, stream>>>(x, Win, bin, h, h0);

    const int aggBlocks  = (NNODES * HDIM + 255) / 256;
    const int spmmBlocks = (int)(((size_t)NEDGES * 32 + 255) / 256);
    for (int i = 0; i < NLAYERS; ++i) {
        float beta = logf(0.5f / (float)(i + 1) + 1.0f);
        k_agg_init  <<<aggBlocks,  256, 0, stream>>>(h, dinv, agg);
        k_spmm      <<<spmmBlocks, 256, 0, stream>>>(esrc, edst, nrm, h, agg);
        k_layer_gemm<<<tileBlocks, 256, 0, stream>>>(agg, h0, Wl + (size_t)i * HDIM * HDIM, h, beta);
    }

    k_out<<<(NNODES + 63) / 64, 256, 0, stream>>>(h, Wout, bout, out);
}